// SampleAndGroup_63196148793614
// MI455X (gfx1250) — compile-verified
//
#include <hip/hip_runtime.h>

// ---------------------------------------------------------------------------
// PointNet++ sample-and-group for MI455X (gfx1250, wave32).
//   B=8, N=16384, C=128, npoint S=1024, NSAMPLE K=32, radius from d_in[1].
//
// Kernels:
//  1) fps_kernel        : 1 workgroup/batch, points in VGPRs, LDS+shfl argmax.
//  2) gather_newxyz     : new_xyz = xyz[fps_idx]  -> d_out[o0]
//  3) ball_query_wmma   : per (batch, 16-centroid tile) wave; -2*c.x dot term
//                         via V_WMMA_F32_16X16X4_F32 (2 tiles/iter for ILP);
//                         ballot masks are wave-uniform SGPRs, row selection
//                         via a branchless cndmask tree (no LDS, no barriers
//                         in the scan loop); ctz-walk appends ascending idx.
//  4) group_kernel      : new_points = points[idx] (float4 rows), grouped_xyz.
// ---------------------------------------------------------------------------

typedef float v2f __attribute__((ext_vector_type(2)));
typedef float v8f __attribute__((ext_vector_type(8)));

#define B_ 8
#define N_ 16384
#define C_ 128
#define S_ 1024
#define K_ 32
#define PPT_ 16           // points per thread in FPS (N_/1024)

// ---------------------------------------------------------------- FPS ------
__global__ __launch_bounds__(1024) void fps_kernel(
    const float* __restrict__ xyz, int* __restrict__ fps_idx)
{
    const int b   = blockIdx.x;
    const int tid = threadIdx.x;
    const float* xb = xyz + (size_t)b * N_ * 3;

    float px[PPT_], py[PPT_], pz[PPT_], dm[PPT_];
#pragma unroll
    for (int j = 0; j < PPT_; ++j) {
        const int n = tid + j * 1024;
        px[j] = xb[n * 3 + 0];
        py[j] = xb[n * 3 + 1];
        pz[j] = xb[n * 3 + 2];
        dm[j] = 1e10f;
    }

    __shared__ float s_red_d[32];
    __shared__ int   s_red_i[32];
    __shared__ float s_c[3];
    __shared__ int   s_far;

    int far = 0;  // reference: far0 = 0
    for (int i = 0; i < S_; ++i) {
        // owner of current `far` broadcasts its coordinates via LDS
        if (tid == (far & 1023)) {
            const int j = far >> 10;
            s_c[0] = px[j]; s_c[1] = py[j]; s_c[2] = pz[j];
        }
        if (tid == 0) fps_idx[b * S_ + i] = far;   // scan emits carry-in far
        __syncthreads();

        const float cx = s_c[0], cy = s_c[1], cz = s_c[2];
        float bestd = -1.0f;
        int   besti = 0;
#pragma unroll
        for (int j = 0; j < PPT_; ++j) {
            const float dx = px[j] - cx, dy = py[j] - cy, dz = pz[j] - cz;
            const float d2 = dx * dx + dy * dy + dz * dz;
            dm[j] = fminf(dm[j], d2);
            if (dm[j] > bestd) { bestd = dm[j]; besti = tid + j * 1024; }
        }
        // wave32 reduce (argmax, prefer lower index on ties)
#pragma unroll
        for (int off = 16; off > 0; off >>= 1) {
            const float od = __shfl_down(bestd, off, 32);
            const int   oi = __shfl_down(besti, off, 32);
            if (od > bestd || (od == bestd && oi < besti)) { bestd = od; besti = oi; }
        }
        if ((tid & 31) == 0) { s_red_d[tid >> 5] = bestd; s_red_i[tid >> 5] = besti; }
        __syncthreads();
        if (tid < 32) {
            bestd = s_red_d[tid]; besti = s_red_i[tid];
#pragma unroll
            for (int off = 16; off > 0; off >>= 1) {
                const float od = __shfl_down(bestd, off, 32);
                const int   oi = __shfl_down(besti, off, 32);
                if (od > bestd || (od == bestd && oi < besti)) { bestd = od; besti = oi; }
            }
            if (tid == 0) s_far = besti;
        }
        __syncthreads();
        far = s_far;
    }
}

// ------------------------------------------------------- gather new_xyz ----
__global__ __launch_bounds__(256) void gather_newxyz(
    const float* __restrict__ xyz, const int* __restrict__ fps_idx,
    float* __restrict__ out0)
{
    const int i = blockIdx.x * blockDim.x + threadIdx.x;  // over B_*S_
    if (i >= B_ * S_) return;
    const int b = i / S_;
    const int n = fps_idx[i];
    const float* p = xyz + ((size_t)b * N_ + n) * 3;
    float* o = out0 + (size_t)i * 3;
    o[0] = p[0]; o[1] = p[1]; o[2] = p[2];
}

// ------------------------------------------------- ball query with WMMA ----
// Uniform 8-entry ballot -> per-lane row mask via branchless select tree.
__device__ __forceinline__ unsigned sel8(const unsigned u[8], int r)
{
    const unsigned s01 = (r & 1) ? u[1] : u[0];
    const unsigned s23 = (r & 1) ? u[3] : u[2];
    const unsigned s45 = (r & 1) ? u[5] : u[4];
    const unsigned s67 = (r & 1) ? u[7] : u[6];
    const unsigned a   = (r & 2) ? s23 : s01;
    const unsigned c   = (r & 2) ? s67 : s45;
    return (r & 4) ? c : a;
}

__global__ __launch_bounds__(32) void ball_query_wmma(
    const float* __restrict__ xyz, const float* __restrict__ newxyz,
    const float* __restrict__ radius_p,
    int* __restrict__ idx_ws, float* __restrict__ idx_outf)
{
    const int b     = blockIdx.y;
    const int sbase = blockIdx.x * 16;
    const int l     = threadIdx.x;           // 0..31
    const float r   = radius_p[0];
    const float r2  = r * r;

    const float* xb = xyz    + (size_t)b * N_ * 3;
    const float* cb = newxyz + ((size_t)b * S_ + sbase) * 3;

    __shared__ float s_c2[16];

    // A fragment: lane l holds A[m = l&15][k0, k0+1], k0 = 2*(l>>4); pad K=3 -> 0
    const int m  = l & 15;
    const int k0 = (l >> 4) * 2;
    const float cx = cb[m * 3 + 0], cy = cb[m * 3 + 1], cz = cb[m * 3 + 2];
    v2f afrag;
    afrag.x = (k0 == 0) ? cx : cz;
    afrag.y = (k0 == 0) ? cy : 0.0f;
    if (l < 16) s_c2[m] = cx * cx + cy * cy + cz * cz;
    __syncthreads();
    float c2r[8];
#pragma unroll
    for (int k = 0; k < 8; ++k) c2r[k] = s_c2[k + 8 * (l >> 4)];

    int cnt = 0, first = 0;
    int*   myidx  = idx_ws   + ((size_t)(b * S_ + sbase + m)) * K_;
    float* myidxf = idx_outf + ((size_t)(b * S_ + sbase + m)) * K_;

    // Scan N in 2x16-column tiles: two WMMAs per iteration for XDL ILP.
    for (int nb = 0; nb < N_; nb += 32) {
        const int n0 = nb + (l & 15);
        const int n1 = n0 + 16;
        const float x0 = xb[n0 * 3 + 0], y0 = xb[n0 * 3 + 1], z0 = xb[n0 * 3 + 2];
        const float x1 = xb[n1 * 3 + 0], y1 = xb[n1 * 3 + 1], z1 = xb[n1 * 3 + 2];
        const float xn2_0 = x0 * x0 + y0 * y0 + z0 * z0;
        const float xn2_1 = x1 * x1 + y1 * y1 + z1 * z1;
        // B fragments: lane l holds B[k0][n], B[k0+1][n] (xyz^T, K=3 padded 0)
        v2f b0, b1;
        b0.x = (l < 16) ? x0 : z0;  b0.y = (l < 16) ? y0 : 0.0f;
        b1.x = (l < 16) ? x1 : z1;  b1.y = (l < 16) ? y1 : 0.0f;

        v8f acc0 = {}, acc1 = {};
        // 8 args: (neg_a, A, neg_b, B, c_mod, C, reuse_a, reuse_b)
        acc0 = __builtin_amdgcn_wmma_f32_16x16x4_f32(
            false, afrag, false, b0, (short)0, acc0, false, false);
        acc1 = __builtin_amdgcn_wmma_f32_16x16x4_f32(
            false, afrag, false, b1, (short)0, acc1, false, false);

        // acc[k] (lane l) = c_{k+8*(l>>4)} . x_n ; ballot bit l -> (row, col)
        unsigned u0[8], u1[8];
#pragma unroll
        for (int k = 0; k < 8; ++k) {
            const float d2a = c2r[k] + xn2_0 - 2.0f * acc0[k];
            const float d2b = c2r[k] + xn2_1 - 2.0f * acc1[k];
            u0[k] = (unsigned)__ballot(d2a < r2);  // bits 0-15: row k, 16-31: row k+8
            u1[k] = (unsigned)__ballot(d2b < r2);
        }

        if (l < 16 && cnt < K_) {
            const unsigned mk0 = sel8(u0, m & 7);
            const unsigned mk1 = sel8(u1, m & 7);
            unsigned rm0 = (m < 8) ? (mk0 & 0xFFFFu) : (mk0 >> 16);
            unsigned rm1 = (m < 8) ? (mk1 & 0xFFFFu) : (mk1 >> 16);
            while (rm0 && cnt < K_) {
                const int nidx = nb + (__ffs(rm0) - 1);
                if (cnt == 0) first = nidx;
                myidx[cnt]  = nidx;
                myidxf[cnt] = (float)nidx;
                ++cnt;
                rm0 &= rm0 - 1;
            }
            while (rm1 && cnt < K_) {
                const int nidx = nb + 16 + (__ffs(rm1) - 1);
                if (cnt == 0) first = nidx;
                myidx[cnt]  = nidx;
                myidxf[cnt] = (float)nidx;
                ++cnt;
                rm1 &= rm1 - 1;
            }
        }

        const bool done = (l >= 16) || (cnt >= K_);
        if (__all(done)) break;
    }

    if (l < 16) {
        const int fill = (cnt > 0) ? first : 0;
        for (int j = cnt; j < K_; ++j) { myidx[j] = fill; myidxf[j] = (float)fill; }
    }
}

// ------------------------------------------------------------- grouping ----
// One block per (b,s): 8 threads per sample copy a 512B points row (float4),
// lanes 0..31 also produce grouped_xyz = xyz[idx] - new_xyz.
__global__ __launch_bounds__(256) void group_kernel(
    const float* __restrict__ xyz, const float* __restrict__ points,
    const int* __restrict__ idx_ws, const float* __restrict__ newxyz,
    float* __restrict__ out_newpoints, float* __restrict__ out_groupedxyz)
{
    const int sid = blockIdx.x;          // 0..B_*S_-1
    const int b   = sid / S_;
    const int t   = threadIdx.x;
    const float* pb = points + (size_t)b * N_ * C_;
    const float* xb = xyz    + (size_t)b * N_ * 3;
    const int*   gi = idx_ws + (size_t)sid * K_;

    __shared__ int   s_idx[K_];
    __shared__ float s_cent[3];
    if (t < K_) s_idx[t]  = gi[t];
    if (t < 3)  s_cent[t] = newxyz[(size_t)sid * 3 + t];
    __syncthreads();

    // new_points: sample k = t>>3, 8 threads stride over 32 float4s of the row
    const int k    = t >> 3;
    const int sub  = t & 7;
    const int pidx = s_idx[k];
    const float4* src = (const float4*)(pb + (size_t)pidx * C_);
    float4*       dst = (float4*)(out_newpoints + ((size_t)sid * K_ + k) * C_);
#pragma unroll
    for (int c = 0; c < 4; ++c)
        dst[c * 8 + sub] = src[c * 8 + sub];

    // grouped_xyz
    if (t < K_) {
        const int pi = s_idx[t];
        float* g = out_groupedxyz + ((size_t)sid * K_ + t) * 3;
        g[0] = xb[pi * 3 + 0] - s_cent[0];
        g[1] = xb[pi * 3 + 1] - s_cent[1];
        g[2] = xb[pi * 3 + 2] - s_cent[2];
    }
}

// --------------------------------------------------------------- launch ----
extern "C" void kernel_launch(void* const* d_in, const int* in_sizes, int n_in,
                              void* d_out, int out_size, void* d_ws, size_t ws_size,
                              hipStream_t stream)
{
    (void)in_sizes; (void)n_in; (void)out_size; (void)ws_size;

    // setup_inputs order: npoint(int), radius(float), xyz, points
    const float* radius = (const float*)d_in[1];
    const float* xyz    = (const float*)d_in[2];
    const float* points = (const float*)d_in[3];
    float* out = (float*)d_out;

    // flat f32 output offsets: new_xyz | new_points | idx(as float) | grouped_xyz
    const size_t o0 = 0;
    const size_t o1 = (size_t)B_ * S_ * 3;                    // 24576
    const size_t o2 = o1 + (size_t)B_ * S_ * K_ * C_;         // 33579008
    const size_t o3 = o2 + (size_t)B_ * S_ * K_;              // 33841152

    int* fps_idx = (int*)d_ws;                 // B_*S_ ints
    int* idx_ws  = fps_idx + (size_t)B_ * S_;  // B_*S_*K_ ints

    fps_kernel<<<B_, 1024, 0, stream>>>(xyz, fps_idx);
    gather_newxyz<<<(B_ * S_ + 255) / 256, 256, 0, stream>>>(xyz, fps_idx, out + o0);

    dim3 gq(S_ / 16, B_);
    ball_query_wmma<<<gq, 32, 0, stream>>>(xyz, out + o0, radius, idx_ws, out + o2);

    group_kernel<<<B_ * S_, 256, 0, stream>>>(xyz, points, idx_ws, out + o0,
                                              out + o1, out + o3);
}